// ResidualGATv2Layer_12549894439331
// MI455X (gfx1250) — compile-verified
//
#include <hip/hip_runtime.h>
#include <hip/hip_bf16.h>

#define NNODES 50000
#define NEDGES 800000
#define INC    128
#define NH     8
#define HC     256
#define NEG_SLOPE 0.2f

typedef float v2f __attribute__((ext_vector_type(2)));
typedef float v8f __attribute__((ext_vector_type(8)));

// Order-preserving float -> uint key (for atomicMax-based segment max).
__device__ __forceinline__ unsigned fkey(float f) {
  unsigned b = __float_as_uint(f);
  return (b & 0x80000000u) ? ~b : (b | 0x80000000u);
}
__device__ __forceinline__ float keyf(unsigned k) {
  unsigned b = (k & 0x80000000u) ? (k & 0x7FFFFFFFu) : ~k;
  return __uint_as_float(b);
}

// ---------------------------------------------------------------------------
// Fused triple GEMM: XL = x@Wl, XR = x@Wr, RS = x@Wp   (fp32 WMMA 16x16x4)
// Block: 256 threads = 8 waves. Each block: 16 rows x 128 cols of the
// combined 768-wide output; wave w owns a 16x16 tile.
// ---------------------------------------------------------------------------
__global__ __launch_bounds__(256)
void gemm3_wmma(const float* __restrict__ x,
                const float* __restrict__ Wl,
                const float* __restrict__ Wr,
                const float* __restrict__ Wp,
                float* __restrict__ XL,
                float* __restrict__ XR,
                float* __restrict__ RS) {
  __shared__ float As[16][132];              // pad: 132 % 64 banks -> no conflict
  const int tid = threadIdx.x;
  const int r0  = blockIdx.x * 16;

  // Cooperative load of 16x128 A tile (8 floats / thread)
  {
    int row = tid >> 4;
    int col = (tid & 15) * 8;
    const float4* s4 = reinterpret_cast<const float4*>(&x[(size_t)(r0 + row) * INC + col]);
    float4 p0 = s4[0];
    float4 p1 = s4[1];
    float* d = &As[row][col];
    d[0]=p0.x; d[1]=p0.y; d[2]=p0.z; d[3]=p0.w;
    d[4]=p1.x; d[5]=p1.y; d[6]=p1.z; d[7]=p1.w;
  }
  __syncthreads();

  const int wave = tid >> 5;
  const int lane = tid & 31;
  const int n    = lane & 15;                // output column within tile / A row
  const int lh   = lane >> 4;                // lane-half selects K pair
  const int gj   = blockIdx.y * 128 + wave * 16 + n;   // global col in [0,768)
  const int wsel = gj >> 8;
  const int lc   = gj & (HC - 1);

  const float* __restrict__ W = (wsel == 0) ? Wl : (wsel == 1 ? Wr : Wp);
  float* __restrict__ Y       = (wsel == 0) ? XL : (wsel == 1 ? XR : RS);

  v8f acc = {0.f, 0.f, 0.f, 0.f, 0.f, 0.f, 0.f, 0.f};
  #pragma unroll 4
  for (int k0 = 0; k0 < INC; k0 += 4) {
    const int ka = k0 + 2 * lh;              // lanes 0-15: K=k0,k0+1; 16-31: K=k0+2,k0+3
    v2f a, b;
    a.x = As[n][ka];
    a.y = As[n][ka + 1];
    b.x = W[(size_t)ka * HC + lc];
    b.y = W[(size_t)(ka + 1) * HC + lc];
    acc = __builtin_amdgcn_wmma_f32_16x16x4_f32(
        /*neg_a=*/false, a, /*neg_b=*/false, b,
        /*c_mod=*/(short)0, acc, /*reuse_a=*/false, /*reuse_b=*/false);
  }

  // C/D layout: VGPR i -> M = i + 8*lane_half, N = lane&15
  const int mb = lh * 8;
  #pragma unroll
  for (int i = 0; i < 8; ++i) {
    Y[(size_t)(r0 + mb + i) * HC + lc] = acc[i];
  }
}

// ---------------------------------------------------------------------------
// Per-edge GATv2 score + segment max.  Wave-per-edge: 32 lanes x 8 channels.
// ---------------------------------------------------------------------------
__global__ __launch_bounds__(256)
void edge_score(const float* __restrict__ XL, const float* __restrict__ XR,
                const int* __restrict__ ei, const float* __restrict__ att,
                float* __restrict__ score, unsigned* __restrict__ mkey) {
  __shared__ float satt[HC];
  satt[threadIdx.x] = att[threadIdx.x];
  __syncthreads();

  const int lane = threadIdx.x & 31;
  const int e = blockIdx.x * 8 + (threadIdx.x >> 5);
  if (e >= NEDGES) return;
  const int src = ei[e];
  const int dst = ei[NEDGES + e];
  const int base = lane * 8;

  const float4* pl = reinterpret_cast<const float4*>(&XL[(size_t)src * HC + base]);
  const float4* pr = reinterpret_cast<const float4*>(&XR[(size_t)dst * HC + base]);
  float4 l0 = pl[0], l1 = pl[1];
  float4 q0 = pr[0], q1 = pr[1];
  float lv[8] = {l0.x, l0.y, l0.z, l0.w, l1.x, l1.y, l1.z, l1.w};
  float rv[8] = {q0.x, q0.y, q0.z, q0.w, q1.x, q1.y, q1.z, q1.w};

  float acc = 0.f;
  #pragma unroll
  for (int i = 0; i < 8; ++i) {
    float v = lv[i] + rv[i];
    v = (v >= 0.f) ? v : NEG_SLOPE * v;      // LeakyReLU(0.2)
    acc += satt[base + i] * v;
  }
  // reduce 4 lanes (one head spans lanes 4h..4h+3)
  acc += __shfl_xor(acc, 1, 32);
  acc += __shfl_xor(acc, 2, 32);
  if ((lane & 3) == 0) {
    const int h = lane >> 2;
    score[(size_t)e * NH + h] = acc;
    atomicMax(&mkey[(size_t)dst * NH + h], fkey(acc));
  }
}

// ---------------------------------------------------------------------------
// ex = exp(score - m[dst]); denom[dst] += ex   (thread per edge*head)
// ---------------------------------------------------------------------------
__global__ __launch_bounds__(256)
void edge_exp(const int* __restrict__ ei, const unsigned* __restrict__ mkey,
              float* __restrict__ score, float* __restrict__ denom) {
  const int idx = blockIdx.x * blockDim.x + threadIdx.x;
  if (idx >= NEDGES * NH) return;
  const int e = idx >> 3;
  const int h = idx & 7;
  const int dst = ei[NEDGES + e];
  const float m = keyf(mkey[(size_t)dst * NH + h]);
  const float ex = __expf(score[idx] - m);
  score[idx] = ex;                           // overwrite in place
  atomicAdd(&denom[(size_t)dst * NH + h], ex);
}

// ---------------------------------------------------------------------------
// out[dst] += (ex/denom[dst]) * XL[src]    (wave-per-edge scatter-add)
// ---------------------------------------------------------------------------
__global__ __launch_bounds__(256)
void edge_aggregate(const float* __restrict__ XL, const int* __restrict__ ei,
                    const float* __restrict__ ex, const float* __restrict__ denom,
                    float* __restrict__ out) {
  const int lane = threadIdx.x & 31;
  const int e = blockIdx.x * 8 + (threadIdx.x >> 5);
  if (e >= NEDGES) return;
  const int src = ei[e];
  const int dst = ei[NEDGES + e];
  const int h = lane >> 2;
  const float alpha = ex[(size_t)e * NH + h] /
                      fmaxf(denom[(size_t)dst * NH + h], 1e-16f);
  const int base = lane * 8;
  const float4* pl = reinterpret_cast<const float4*>(&XL[(size_t)src * HC + base]);
  float4 l0 = pl[0], l1 = pl[1];
  float lv[8] = {l0.x, l0.y, l0.z, l0.w, l1.x, l1.y, l1.z, l1.w};
  float* o = &out[(size_t)dst * HC + base];
  #pragma unroll
  for (int i = 0; i < 8; ++i) atomicAdd(&o[i], alpha * lv[i]);
}

// ---------------------------------------------------------------------------
// out = gelu(out + bias + res + bp)   (exact erf GELU), in place
// ---------------------------------------------------------------------------
__global__ __launch_bounds__(256)
void finish_gelu(const float* __restrict__ RS, const float* __restrict__ bias,
                 const float* __restrict__ bp, float* __restrict__ out) {
  const int idx = blockIdx.x * blockDim.x + threadIdx.x;
  if (idx >= NNODES * HC) return;
  const int j = idx & (HC - 1);
  const float v = out[idx] + bias[j] + RS[idx] + bp[j];
  out[idx] = 0.5f * v * (1.0f + erff(v * 0.70710678118654752f));
}

extern "C" void kernel_launch(void* const* d_in, const int* in_sizes, int n_in,
                              void* d_out, int out_size, void* d_ws, size_t ws_size,
                              hipStream_t stream) {
  const float* x    = (const float*)d_in[0];
  const int*   ei   = (const int*)d_in[1];
  const float* Wl   = (const float*)d_in[2];
  const float* Wr   = (const float*)d_in[3];
  const float* att  = (const float*)d_in[4];
  const float* bias = (const float*)d_in[5];
  const float* Wp   = (const float*)d_in[6];
  const float* bp   = (const float*)d_in[7];
  float* out = (float*)d_out;

  // Workspace layout (~183 MB): XL | XR | RS | score/ex | mkey | denom
  float*    XL    = (float*)d_ws;
  float*    XR    = XL + (size_t)NNODES * HC;
  float*    RS    = XR + (size_t)NNODES * HC;
  float*    score = RS + (size_t)NNODES * HC;
  unsigned* mkey  = (unsigned*)(score + (size_t)NEDGES * NH);
  float*    denom = (float*)(mkey + (size_t)NNODES * NH);

  hipMemsetAsync(out,   0, sizeof(float)    * (size_t)NNODES * HC, stream);
  hipMemsetAsync(mkey,  0, sizeof(unsigned) * (size_t)NNODES * NH, stream);
  hipMemsetAsync(denom, 0, sizeof(float)    * (size_t)NNODES * NH, stream);

  dim3 g1(NNODES / 16, (3 * HC) / 128);               // 3125 x 6
  gemm3_wmma<<<g1, 256, 0, stream>>>(x, Wl, Wr, Wp, XL, XR, RS);

  edge_score<<<NEDGES / 8, 256, 0, stream>>>(XL, XR, ei, att, score, mkey);
  edge_exp<<<(NEDGES * NH) / 256, 256, 0, stream>>>(ei, mkey, score, denom);
  edge_aggregate<<<NEDGES / 8, 256, 0, stream>>>(XL, ei, score, denom, out);
  finish_gelu<<<(NNODES * HC) / 256, 256, 0, stream>>>(RS, bias, bp, out);
}